// EquivariantSubSampling_44117904064575
// MI455X (gfx1250) — compile-verified
//
#include <hip/hip_runtime.h>

#define TILE_PAD 33

__global__ __launch_bounds__(256) void eqsub_gather_kernel(
    const float* __restrict__ x,
    const int*   __restrict__ p,
    float*       __restrict__ out)
{
    // Padded 32x32 fp32 tile: row stride 33 dwords -> conflict-free column reads
    __shared__ float tile[32][TILE_PAD];

    const int blk = blockIdx.x;      // = b*128 + k, k = output channel
    const int b   = blk >> 7;
    const int k   = blk & 127;
    const int cp  = k >> 1;          // channel group (0..63)
    const int rp  = k & 1;           // fiber-subsample slot (0..1)

    const int p0 = p[b * 3 + 0];     // uniform per block
    const int p1 = p[b * 3 + 1];
    const int p2 = p[b * 3 + 2];

    const int f  = p2 + 2 * rp;      // fiber index within group of 4
    const int ch = (cp << 2) + f;    // input channel (0..255)

    const int t = threadIdx.x;

    // base of x[b][ch][.][.] : plane is 64*64 floats
    const float* xp = x + (((size_t)b * 256 + (size_t)ch) << 12);

    // LDS byte offset of the tile (flat LDS address low 32 bits == LDS offset)
    const unsigned lds_base = (unsigned)(size_t)(&tile[0][0]);

    // ---- phase 1: async copy stride-2 32x32 tile, global -> LDS (no VGPR trip) ----
    #pragma unroll
    for (int it = 0; it < 4; ++it) {
        const int e = (it << 8) + t;          // 0..1023
        const int i = e >> 5;                 // tile row
        const int j = e & 31;                 // tile col
        const float* ga = xp + (size_t)(((p0 + 2 * i) << 6) + (p1 + 2 * j));
        const unsigned lofs = lds_base + 4u * (unsigned)(i * TILE_PAD + j);
        asm volatile("global_load_async_to_lds_b32 %0, %1, off"
                     :: "v"(lofs), "v"(ga)
                     : "memory");
    }
    asm volatile("s_wait_asynccnt 0x0" ::: "memory");
    __syncthreads();

    // ---- phase 2: async copy LDS -> global, applying identity or 90-deg rotation ----
    float* op = out + ((size_t)blk << 10);    // 32*32 per (b,k)

    if (p2 == 0) {
        #pragma unroll
        for (int it = 0; it < 4; ++it) {
            const int e = (it << 8) + t;
            const int u = e >> 5;
            const int v = e & 31;
            const float* gd = op + e;                                   // coalesced
            const unsigned lofs = lds_base + 4u * (unsigned)(u * TILE_PAD + v);
            asm volatile("global_store_async_from_lds_b32 %0, %1, off"
                         :: "v"(gd), "v"(lofs)
                         : "memory");
        }
    } else {
        #pragma unroll
        for (int it = 0; it < 4; ++it) {
            const int e = (it << 8) + t;
            const int u = e >> 5;
            const int v = e & 31;
            const float* gd = op + e;                                   // coalesced
            const unsigned lofs =
                lds_base + 4u * (unsigned)(((32 - v) & 31) * TILE_PAD + u); // conflict-free
            asm volatile("global_store_async_from_lds_b32 %0, %1, off"
                         :: "v"(gd), "v"(lofs)
                         : "memory");
        }
    }
    asm volatile("s_wait_asynccnt 0x0" ::: "memory");
}

extern "C" void kernel_launch(void* const* d_in, const int* in_sizes, int n_in,
                              void* d_out, int out_size, void* d_ws, size_t ws_size,
                              hipStream_t stream) {
    const float* x   = (const float*)d_in[0];   // (16,256,64,64) fp32
    const int*   p   = (const int*)d_in[1];     // (16,3) int32
    float*       out = (float*)d_out;           // (16,128,32,32) fp32

    (void)in_sizes; (void)n_in; (void)out_size; (void)d_ws; (void)ws_size;

    dim3 grid(16 * 128);   // one block per (batch, output channel)
    dim3 block(256);       // 8 waves (wave32)
    hipLaunchKernelGGL(eqsub_gather_kernel, grid, block, 0, stream, x, p, out);
}